// NNFBP_10264971837719
// MI455X (gfx1250) — compile-verified
//
#include <hip/hip_runtime.h>
#include <math.h>

// ---------------------------------------------------------------------------
// Geometry constants (match reference)
// ---------------------------------------------------------------------------
#define AANG 400
#define DDET 512
#define NN   256
#define KS   11
// DS = 4/512 = 0.0078125 ; 1/DS = 128 ; u = S*128 + 255.5
// For this geometry u is provably in [75, 436], so no detector bounds checks.
#define U_OFF    255.5f
#define INV_DS   128.0f
#define DPHI_F   0.007853981633974483f   // pi/400
#define PI_F     3.14159265358979323846f

typedef float v2f __attribute__((ext_vector_type(2)));
typedef float v8f __attribute__((ext_vector_type(8)));

// ---------------------------------------------------------------------------
// Kernel 0: per-angle trig table: su = sin(phi)/DS, cu = cos(phi)/DS
// ---------------------------------------------------------------------------
__global__ void trig_kernel(float* __restrict__ su_t, float* __restrict__ cu_t) {
    int a = blockIdx.x * blockDim.x + threadIdx.x;
    if (a < AANG) {
        float phi = (a + 0.5f) * (PI_F / (float)AANG);
        float s, c;
        __sincosf(phi, &s, &c);
        su_t[a] = s * INV_DS;
        cu_t[a] = c * INV_DS;
    }
}

// ---------------------------------------------------------------------------
// Kernel 1: causal detector conv, 1 -> 32 channels, K=11.
// Output layout channel-last: y[(a*512 + d)*32 + c]; lane = c.
// ---------------------------------------------------------------------------
__global__ void detconv_kernel(const float* __restrict__ x,
                               const float* __restrict__ wdet,
                               const float* __restrict__ bdet,
                               float* __restrict__ y) {
    int idx = blockIdx.x * 256 + threadIdx.x;     // exactly 400*512*32 threads
    int c = idx & 31;
    int d = (idx >> 5) & (DDET - 1);
    int a = idx >> 14;
    float s = bdet[c];
    const float* xr = x + a * DDET;
#pragma unroll
    for (int k = 0; k < KS; ++k) {
        int dd = d - (KS - 1) + k;
        float xv = (dd >= 0) ? xr[dd] : 0.0f;
        s += wdet[c * KS + k] * xv;
    }
    y[idx] = s;
}

// ---------------------------------------------------------------------------
// Kernel 2: backprojection. lane = channel; each wave owns one 16-pixel row
// segment; block = 8 waves = 8x16 pixel tile. Both interpolation taps are
// coalesced 128B loads from a wave-uniform row base (saddr form), second tap
// at a +128B immediate offset. No bounds checks (u in [75,436] always).
// vol layout channel-last: vol[(i*256 + j)*32 + c]
// ---------------------------------------------------------------------------
__global__ void bp_kernel(const float* __restrict__ y,
                          const float* __restrict__ su_t,
                          const float* __restrict__ cu_t,
                          float* __restrict__ vol) {
    const int lane = threadIdx.x & 31;
    const int wave = threadIdx.x >> 5;
    const int bi = blockIdx.x >> 4;    // 0..31 (row tile)
    const int bj = blockIdx.x & 15;    // 0..15 (col tile)
    const int i  = bi * 8 + wave;
    const int j0 = bj * 16;

    const float xi = -1.0f + (i + 0.5f) * (2.0f / (float)NN);
    float xj[16];
#pragma unroll
    for (int p = 0; p < 16; ++p)
        xj[p] = -1.0f + (j0 + p + 0.5f) * (2.0f / (float)NN);

    float acc[16];
#pragma unroll
    for (int p = 0; p < 16; ++p) acc[p] = 0.0f;

    for (int a = 0; a < AANG; ++a) {
        const float su = su_t[a];
        const float cu = cu_t[a];
        const float* __restrict__ ya = y + a * (DDET * 32);  // wave-uniform
        const float base = fmaf(-xi, su, U_OFF);
#pragma unroll
        for (int p = 0; p < 16; ++p) {
            float u   = fmaf(xj[p], cu, base);   // fractional detector index
            float k0f = floorf(u);
            float w   = u - k0f;
            int   off = (((int)k0f) << 5) + lane;  // 32-bit element offset
            float g0 = ya[off];
            float g1 = ya[off + 32];
            acc[p] += (1.0f - w) * g0 + w * g1;
        }
    }

#pragma unroll
    for (int p = 0; p < 16; ++p)
        vol[(i * NN + (j0 + p)) * 32 + lane] = acc[p] * DPHI_F;
}

// ---------------------------------------------------------------------------
// Kernel 3: repack W1 [32,32,3,3] (OIHW) into WMMA A-fragment order.
// K dimension order: K = t*32 + c  (t = tap = ki*3+kj, c = input channel).
// Entry (mt, kc, lane) holds the float2 for lane's A VGPR pair:
//   m = lane&15, koff = (lane>=16)?2:0, K = kc*4+koff  -> values (K, K+1).
// ---------------------------------------------------------------------------
__global__ void prep_w1_kernel(const float* __restrict__ w1,
                               float* __restrict__ w1f) {
    int idx = blockIdx.x * blockDim.x + threadIdx.x;   // < 2*72*32 = 4608
    if (idx >= 2 * 72 * 32) return;
    int lane = idx & 31;
    int kc   = (idx >> 5) % 72;
    int mt   = idx / (72 * 32);
    int m    = lane & 15;
    int koff = (lane >> 4) << 1;          // 0 or 2
    int K    = kc * 4 + koff;
    int t    = K >> 5;                    // tap 0..8
    int c    = K & 31;                    // input channel (c%4 in {0,2})
    int o    = mt * 16 + m;
    int ki = t / 3, kj = t % 3;
    float a0 = w1[((o * 32 + c) * 3 + ki) * 3 + kj];
    float a1 = w1[((o * 32 + (c + 1)) * 3 + ki) * 3 + kj];
    w1f[idx * 2 + 0] = a0;
    w1f[idx * 2 + 1] = a1;
}

// ---------------------------------------------------------------------------
// Kernel 4: conv1 (32->32, taps at offsets -2..0) + bias + sigmoid via
// V_WMMA_F32_16X16X4_F32. Each wave: 64 pixels (4 N-tiles of one row) and
// two 16-channel M-tiles -> 8 accumulators, 9 taps x 8 quads x 8 = 576 WMMA.
// Per tap: one address/mask setup; per quad all loads use immediate offsets,
// and each A fragment is reused by 4 WMMAs.
// Input: vol channel-last [(i*256+j)*32+c]; output s1 plane-major [c][i][j].
// ---------------------------------------------------------------------------
__global__ void conv1_wmma_kernel(const float* __restrict__ vol,
                                  const float* __restrict__ w1f,
                                  const float* __restrict__ b1,
                                  float* __restrict__ s1) {
    const int lane   = threadIdx.x & 31;
    const int wave   = threadIdx.x >> 5;
    const int gw     = blockIdx.x * 8 + wave;   // 0..1023
    const int i      = gw >> 2;                 // output row 0..255
    const int j0     = (gw & 3) << 6;           // output col base (0,64,128,192)
    const int n      = lane & 15;               // pixel within N-tile
    const int hiHalf = lane >> 4;               // 0/1 -> K(+0,+1) vs K(+2,+3)

    const v2f* __restrict__ w1fv = (const v2f*)w1f;
    const v2f* __restrict__ volv = (const v2f*)vol;

    v8f acc[2][4];
#pragma unroll
    for (int mt = 0; mt < 2; ++mt)
#pragma unroll
        for (int nt = 0; nt < 4; ++nt)
            acc[mt][nt] = (v8f){0.f, 0.f, 0.f, 0.f, 0.f, 0.f, 0.f, 0.f};

    int dy = -2, dx = -2;
#pragma unroll 1
    for (int t = 0; t < 9; ++t) {
        const int ii  = i + dy;
        const int iic = ii < 0 ? 0 : ii;

        // Per-tile base pointers and zero-pad masks (computed once per tap).
        const v2f* bp[4];
        float msk[4];
#pragma unroll
        for (int nt = 0; nt < 4; ++nt) {
            const int jj  = j0 + nt * 16 + n + dx;
            const int jjc = jj < 0 ? 0 : jj;
            msk[nt] = (ii >= 0 && jj >= 0) ? 1.0f : 0.0f;
            bp[nt]  = volv + ((iic * NN + jjc) * 16 + hiHalf);
        }
        const v2f* aw0 = w1fv + (t * 256 + lane);          // M-tile 0
        const v2f* aw1 = w1fv + (2304 + t * 256 + lane);   // M-tile 1

#pragma unroll
        for (int q = 0; q < 8; ++q) {
            v2f a0 = aw0[q * 32];          // +256B per quad (imm offset)
            v2f a1 = aw1[q * 32];
            v2f b0 = bp[0][q * 2] * msk[0];  // +16B per quad (imm offset)
            v2f b1f = bp[1][q * 2] * msk[1];
            v2f b2f = bp[2][q * 2] * msk[2];
            v2f b3f = bp[3][q * 2] * msk[3];

            acc[0][0] = __builtin_amdgcn_wmma_f32_16x16x4_f32(false, a0, false, b0,  (short)0, acc[0][0], false, false);
            acc[0][1] = __builtin_amdgcn_wmma_f32_16x16x4_f32(false, a0, false, b1f, (short)0, acc[0][1], false, false);
            acc[0][2] = __builtin_amdgcn_wmma_f32_16x16x4_f32(false, a0, false, b2f, (short)0, acc[0][2], false, false);
            acc[0][3] = __builtin_amdgcn_wmma_f32_16x16x4_f32(false, a0, false, b3f, (short)0, acc[0][3], false, false);
            acc[1][0] = __builtin_amdgcn_wmma_f32_16x16x4_f32(false, a1, false, b0,  (short)0, acc[1][0], false, false);
            acc[1][1] = __builtin_amdgcn_wmma_f32_16x16x4_f32(false, a1, false, b1f, (short)0, acc[1][1], false, false);
            acc[1][2] = __builtin_amdgcn_wmma_f32_16x16x4_f32(false, a1, false, b2f, (short)0, acc[1][2], false, false);
            acc[1][3] = __builtin_amdgcn_wmma_f32_16x16x4_f32(false, a1, false, b3f, (short)0, acc[1][3], false, false);
        }
        // advance tap (dy,dx) over {-2,-1,0}^2 without div/mod
        if (dx == 0) { dx = -2; ++dy; } else { ++dx; }
    }

    // Epilogue: D layout -> lane: N = lane&15 ; M = r + 8*(lane>=16)
#pragma unroll
    for (int r = 0; r < 8; ++r) {
        const int chL = r + (hiHalf << 3);
#pragma unroll
        for (int mt = 0; mt < 2; ++mt) {
            const int ch = chL + mt * 16;
            const float bch = b1[ch];
            float* __restrict__ plane = s1 + ch * (NN * NN) + i * NN;
#pragma unroll
            for (int nt = 0; nt < 4; ++nt) {
                float v = acc[mt][nt][r] + bch;
                plane[j0 + nt * 16 + n] = 1.0f / (1.0f + __expf(-v));
            }
        }
    }
}

// ---------------------------------------------------------------------------
// Kernel 5: conv2 (32->1, taps -2..0) + sigmoid. One thread per pixel;
// all loads coalesced along j, s1 is L2-resident (8 MB).
// ---------------------------------------------------------------------------
__global__ void conv2_kernel(const float* __restrict__ s1,
                             const float* __restrict__ w2,
                             const float* __restrict__ b2,
                             float* __restrict__ out) {
    int idx = blockIdx.x * 256 + threadIdx.x;   // 65536 threads
    int j = idx & (NN - 1);
    int i = idx >> 8;
    float sum = b2[0];
    for (int c = 0; c < 32; ++c) {
        const float* __restrict__ pc = s1 + c * (NN * NN);
#pragma unroll
        for (int ki = 0; ki < 3; ++ki) {
            int ii = i + ki - 2;
#pragma unroll
            for (int kj = 0; kj < 3; ++kj) {
                int jj = j + kj - 2;
                if (ii >= 0 && jj >= 0)
                    sum += w2[(c * 3 + ki) * 3 + kj] * pc[ii * NN + jj];
            }
        }
    }
    out[idx] = 1.0f / (1.0f + __expf(-sum));
}

// ---------------------------------------------------------------------------
// Launcher. Workspace layout (floats):
//   y     : 400*512*32 = 6,553,600
//   vol   : 256*256*32 = 2,097,152
//   s1    : 32*256*256 = 2,097,152
//   w1f   : 2*72*32*2  =     9,216
//   su/cu :   400 each
// Total ~43 MB.
// ---------------------------------------------------------------------------
extern "C" void kernel_launch(void* const* d_in, const int* in_sizes, int n_in,
                              void* d_out, int out_size, void* d_ws, size_t ws_size,
                              hipStream_t stream) {
    const float* x     = (const float*)d_in[0];  // [1,1,400,512]
    const float* w_det = (const float*)d_in[1];  // [32,1,1,11]
    const float* b_det = (const float*)d_in[2];  // [32]
    const float* w1    = (const float*)d_in[3];  // [32,32,3,3]
    const float* b1    = (const float*)d_in[4];  // [32]
    const float* w2    = (const float*)d_in[5];  // [1,32,3,3]
    const float* b2    = (const float*)d_in[6];  // [1]
    float* out = (float*)d_out;                  // [1,1,256,256]

    float* ws   = (float*)d_ws;
    float* y    = ws;                     // 6,553,600 floats
    float* vol  = y   + 6553600;          // 2,097,152 floats
    float* s1   = vol + 2097152;          // 2,097,152 floats
    float* w1f  = s1  + 2097152;          // 9,216 floats
    float* su_t = w1f + 9216;             // 400 floats
    float* cu_t = su_t + 400;             // 400 floats

    trig_kernel<<<2, 256, 0, stream>>>(su_t, cu_t);
    detconv_kernel<<<25600, 256, 0, stream>>>(x, w_det, b_det, y);
    prep_w1_kernel<<<18, 256, 0, stream>>>(w1, w1f);
    bp_kernel<<<512, 256, 0, stream>>>(y, su_t, cu_t, vol);
    conv1_wmma_kernel<<<128, 256, 0, stream>>>(vol, w1f, b1, s1);
    conv2_kernel<<<256, 256, 0, stream>>>(s1, w2, b2, out);
}